// FrameSimilarity_37134287241739
// MI455X (gfx1250) — compile-verified
//
#include <hip/hip_runtime.h>

typedef __attribute__((ext_vector_type(2))) float v2f;
typedef __attribute__((ext_vector_type(8))) float v8f;

#define T_DIM  2048
#define F_INN  2048
#define D_DIM  128
#define BATCH  4
#define WIN    101
#define HWIN   50

// ---------------------------------------------------------------------------
// Kernel 0: Wt[c][k] = W[k][c]  (128 x 2048), plus zero the dummy row used by
// the band kernel for out-of-range columns. 1 MB total -- negligible.
// ---------------------------------------------------------------------------
__global__ __launch_bounds__(256) void transpose_w_kernel(
    const float* __restrict__ W, float* __restrict__ Wt,
    float* __restrict__ zrow)
{
  const int idx = blockIdx.x * 256 + threadIdx.x;   // 0 .. 2048*128-1
  const int r = idx >> 7;                            // k index, 0..2047
  const int c = idx & 127;                           // d index, 0..127
  Wt[(size_t)c * F_INN + r] = W[idx];
  if (blockIdx.x == 0 && threadIdx.x < D_DIM) zrow[threadIdx.x] = 0.f;
}

// ---------------------------------------------------------------------------
// Kernel 1: p = normalize_rows(x @ W + b). One wave per 16-row tile, 8 N-tiles
// in v8f accumulators. A and B fragments are both contiguous float2 loads
// (B from transposed W). Software double-buffered so next-k loads are in
// flight while current-k WMMAs execute.
// ---------------------------------------------------------------------------
__global__ __launch_bounds__(128) void gemm_norm_kernel(
    const float* __restrict__ x, const float* __restrict__ Wt,
    const float* __restrict__ bias, float* __restrict__ p)
{
  const int wave = (blockIdx.x * blockDim.x + threadIdx.x) >> 5;
  const int lane = threadIdx.x & 31;
  const int half = lane >> 4;        // 0: lanes 0-15, 1: lanes 16-31
  const int l16  = lane & 15;
  const int row0 = wave * 16;

  v8f acc[8] = {};

  // Per-lane fragment base pointers (include the half*2 K-offset).
  const float* xrow = x + (size_t)(row0 + l16) * F_INN + half * 2;
  const float* wrow[8];
#pragma unroll
  for (int n = 0; n < 8; ++n)
    wrow[n] = Wt + (size_t)(n * 16 + l16) * F_INN + half * 2;

  // prologue: fragments for k = 0
  v2f a_cur = *(const v2f*)(xrow);
  v2f b_cur[8];
#pragma unroll
  for (int n = 0; n < 8; ++n) b_cur[n] = *(const v2f*)(wrow[n]);

  for (int k = 0; k < F_INN; k += 4) {
    const int kn = k + 4;
    v2f a_nxt = {};
    v2f b_nxt[8] = {};
    if (kn < F_INN) {                       // uniform branch, EXEC untouched
      a_nxt = *(const v2f*)(xrow + kn);
#pragma unroll
      for (int n = 0; n < 8; ++n) b_nxt[n] = *(const v2f*)(wrow[n] + kn);
    }
    __builtin_prefetch(xrow + k + 256, 0, 0);   // stream x ~1KB ahead

#pragma unroll
    for (int n = 0; n < 8; ++n)
      acc[n] = __builtin_amdgcn_wmma_f32_16x16x4_f32(
          false, a_cur, false, b_cur[n], (short)0, acc[n], false, false);

    a_cur = a_nxt;
#pragma unroll
    for (int n = 0; n < 8; ++n) b_cur[n] = b_nxt[n];
  }

  // bias add
#pragma unroll
  for (int n = 0; n < 8; ++n) {
    const float bv = bias[n * 16 + l16];
#pragma unroll
    for (int j = 0; j < 8; ++j) acc[n][j] += bv;
  }

  // per-row L2 normalization (row of VGPR j is row0 + j + 8*half; its 128
  // elements live in the 16 lanes of this half across the 8 N-tiles).
#pragma unroll
  for (int j = 0; j < 8; ++j) {
    float s = 0.f;
#pragma unroll
    for (int n = 0; n < 8; ++n) s += acc[n][j] * acc[n][j];
    s += __shfl_xor(s, 1, 32);
    s += __shfl_xor(s, 2, 32);
    s += __shfl_xor(s, 4, 32);
    s += __shfl_xor(s, 8, 32);
    const float scale = 1.0f / fmaxf(sqrtf(s), 1e-12f);
#pragma unroll
    for (int n = 0; n < 8; ++n) acc[n][j] *= scale;
  }

#pragma unroll
  for (int j = 0; j < 8; ++j) {
    const int r = row0 + j + 8 * half;
#pragma unroll
    for (int n = 0; n < 8; ++n)
      p[(size_t)r * D_DIM + n * 16 + l16] = acc[n][j];
  }
}

// ---------------------------------------------------------------------------
// Kernel 2: banded P @ P^T. One wave per (batch, 16-row tile); the +-50 band
// spans 9 column tiles. Out-of-range columns read a zeroed dummy row, making
// the loop branch-free and reproducing the reference's zero padding. Every
// output element is written exactly once. Double-buffered like kernel 1.
// ---------------------------------------------------------------------------
__global__ __launch_bounds__(128) void band_sim_kernel(
    const float* __restrict__ p, const float* __restrict__ zrow,
    float* __restrict__ out)
{
  const int wave = (blockIdx.x * blockDim.x + threadIdx.x) >> 5;
  const int lane = threadIdx.x & 31;
  const int half = lane >> 4;
  const int l16  = lane & 15;
  const int b    = wave >> 7;          // 128 row-tiles per batch
  const int mt   = wave & 127;
  const int t0   = mt * 16;

  const float* pb = p + (size_t)b * T_DIM * D_DIM;

  // k-invariant per-lane column base pointers (invalid -> zero row)
  const float* bbase[9];
#pragma unroll
  for (int ct = 0; ct < 9; ++ct) {
    const int c = t0 + (ct - 4) * 16 + l16;
    bbase[ct] = ((unsigned)c < (unsigned)T_DIM)
                    ? (pb + (size_t)c * D_DIM + half * 2)
                    : (zrow + half * 2);
  }
  const float* arow = pb + (size_t)(t0 + l16) * D_DIM + half * 2;

  v8f acc[9] = {};

  v2f a_cur = *(const v2f*)(arow);
  v2f b_cur[9];
#pragma unroll
  for (int ct = 0; ct < 9; ++ct) b_cur[ct] = *(const v2f*)(bbase[ct]);

  for (int k = 0; k < D_DIM; k += 4) {
    const int kn = k + 4;
    v2f a_nxt = {};
    v2f b_nxt[9] = {};
    if (kn < D_DIM) {                       // uniform branch
      a_nxt = *(const v2f*)(arow + kn);
#pragma unroll
      for (int ct = 0; ct < 9; ++ct) b_nxt[ct] = *(const v2f*)(bbase[ct] + kn);
    }

#pragma unroll
    for (int ct = 0; ct < 9; ++ct)
      acc[ct] = __builtin_amdgcn_wmma_f32_16x16x4_f32(
          false, a_cur, false, b_cur[ct], (short)0, acc[ct], false, false);

    a_cur = a_nxt;
#pragma unroll
    for (int ct = 0; ct < 9; ++ct) b_cur[ct] = b_nxt[ct];
  }

  // scatter the band: G[t][c] -> out[b][t][w], w = c - t + 50
#pragma unroll
  for (int ct = 0; ct < 9; ++ct) {
    const int c = t0 + (ct - 4) * 16 + l16;
#pragma unroll
    for (int j = 0; j < 8; ++j) {
      const int t = t0 + j + 8 * half;
      const int w = c - t + HWIN;
      if (w >= 0 && w < WIN) {
        out[((size_t)b * T_DIM + t) * WIN + w] = acc[ct][j];
      }
    }
  }
}

extern "C" void kernel_launch(void* const* d_in, const int* in_sizes, int n_in,
                              void* d_out, int out_size, void* d_ws, size_t ws_size,
                              hipStream_t stream) {
  const float* x  = (const float*)d_in[0];   // (4, 2048, 2048) f32
  const float* W  = (const float*)d_in[1];   // (2048, 128)     f32
  const float* bv = (const float*)d_in[2];   // (128,)          f32
  float* out = (float*)d_out;                // (4, 2048, 101)  f32

  float* p    = (float*)d_ws;                // 8192*128 f32  = 4 MB
  float* Wt   = p + (size_t)8192 * D_DIM;    // 128*2048 f32  = 1 MB
  float* zrow = Wt + (size_t)D_DIM * F_INN;  // 128 f32       = 512 B

  // Wt + zero-row init (stream-ordered before consumers).
  transpose_w_kernel<<<(F_INN * D_DIM) / 256, 256, 0, stream>>>(W, Wt, zrow);
  // 8192 rows / 16 per wave = 512 waves; 4 waves per 128-thread block.
  gemm_norm_kernel<<<128, 128, 0, stream>>>(x, Wt, bv, p);
  // 4 batches * 128 row-tiles = 512 waves.
  band_sim_kernel<<<128, 128, 0, stream>>>(p, zrow, out);
}